// TritonRnntLoss_22497038696572
// MI455X (gfx1250) — compile-verified
//
#include <hip/hip_runtime.h>
#include <stdint.h>
#include <stddef.h>

// Problem constants (from reference): B=4, T=256, U=64, V=1024, BLANK=1023
#define Bb 4
#define Tt 256
#define Uu 64
#define Vv 1024
#define U1 65
#define BLANKI 1023
#define ROWS_PER_BLOCK 8
#define NROWS (Bb * Tt * U1) // 66560 rows, divisible by 8

typedef float        v4f __attribute__((ext_vector_type(4)));
typedef unsigned int v4u __attribute__((ext_vector_type(4)));
typedef int          v4i __attribute__((ext_vector_type(4)));
typedef int          v8i __attribute__((ext_vector_type(8)));

#define INV_LN2f 1.44269504088896340736f
#define LN2f     0.69314718055994530942f

#if defined(__AMDGCN__) && __has_builtin(__builtin_amdgcn_tensor_load_to_lds) && __has_builtin(__builtin_amdgcn_s_wait_tensorcnt)
#define USE_TDM 1
#else
#define USE_TDM 0
#endif

__device__ __forceinline__ float wave_red_max(float v) {
#pragma unroll
  for (int d = 16; d >= 1; d >>= 1) v = fmaxf(v, __shfl_xor(v, d, 32));
  return v;
}
__device__ __forceinline__ float wave_red_sum(float v) {
#pragma unroll
  for (int d = 16; d >= 1; d >>= 1) v += __shfl_xor(v, d, 32);
  return v;
}

// -------- Phase 1: per-(b,t,u) logsumexp over V=1024; emit lp_blank, lp_label --------
// One wave per row; 8 rows per 256-thread block. TDM streams the 8 rows (32KB)
// into LDS, then waves reduce from LDS (ds_load_b128). Fallback: NT global loads.
__global__ __launch_bounds__(256) void rnnt_rowlse_kernel(
    const float* __restrict__ acts, const int* __restrict__ labels,
    float* __restrict__ lpB, float* __restrict__ lpL)
{
#if USE_TDM
  __shared__ float smem[ROWS_PER_BLOCK * Vv];
#endif
  const int rowBase = blockIdx.x * ROWS_PER_BLOCK;
  const int lane = threadIdx.x & 31;
  const int wave = threadIdx.x >> 5;

#if USE_TDM
  if (threadIdx.x < 32) { // wave 0 issues one TDM 1-D tile: 8 rows = 8192 f32 = 32KB
    const uint64_t gaddr = (uint64_t)(uintptr_t)(acts + (size_t)rowBase * Vv);
    const uint32_t laddr = (uint32_t)(uintptr_t)(&smem[0]); // low 32 bits = LDS byte offset
    const uint32_t nelem = ROWS_PER_BLOCK * Vv;             // 8192 (fits tile_dim0 u16)
    v4u g0;
    g0[0] = 1u;                                            // count=1 (valid), no gather
    g0[1] = laddr;                                         // lds_addr (bytes)
    g0[2] = (uint32_t)gaddr;                               // global_addr[31:0]
    g0[3] = (uint32_t)((gaddr >> 32) & 0x01FFFFFFu) | (2u << 30); // addr[56:32] | type=2
    v8i g1;
    g1[0] = (int)(2u << 16);                               // wg_mask=0 (no cluster), data_size=4B
    g1[1] = (int)((nelem & 0xFFFFu) << 16);                // tensor_dim0[15:0] @bits63:48
    g1[2] = (int)((nelem >> 16) & 0xFFFFu) | (int)(1u << 16); // tensor_dim0[31:16], tensor_dim1=1
    g1[3] = (int)((nelem & 0xFFFFu) << 16);                // tile_dim0=8192 @bits127:112
    g1[4] = 1;                                             // tile_dim1=1, tile_dim2=0
    g1[5] = (int)nelem;                                    // tensor_dim0_stride[31:0]
    g1[6] = 0;                                             // stride0[47:32]=0, stride1 lo=0
    g1[7] = 0;
    v4i gz = {0, 0, 0, 0};
#if defined(__clang_major__) && __clang_major__ >= 23
    v8i gz8 = {0, 0, 0, 0, 0, 0, 0, 0};
    __builtin_amdgcn_tensor_load_to_lds(g0, g1, gz, gz, gz8, 0);
#else
    __builtin_amdgcn_tensor_load_to_lds(g0, g1, gz, gz, 0);
#endif
    __builtin_amdgcn_s_wait_tensorcnt((short)0);
  }
  __syncthreads();
  const float* __restrict__ src = &smem[wave * Vv];
#else
  const float* __restrict__ src = acts + (size_t)(rowBase + wave) * Vv;
#endif

  const int row = rowBase + wave;
  const int b = row / (Tt * U1);
  const int rem = row - b * (Tt * U1);
  const int t = rem / U1;
  const int u = rem - t * U1;

  // Row resident in 32 VGPRs/lane: 8 x b128 per lane, coalesced.
  v4f vv[8];
  float m = -3.402823466e+38f;
#pragma unroll
  for (int k = 0; k < 8; ++k) {
#if USE_TDM
    vv[k] = ((const v4f*)src)[lane + 32 * k];
#else
    vv[k] = __builtin_nontemporal_load(((const v4f*)src) + lane + 32 * k);
#endif
    m = fmaxf(m, fmaxf(fmaxf(vv[k][0], vv[k][1]), fmaxf(vv[k][2], vv[k][3])));
  }
  m = wave_red_max(m);

  const float m2 = m * INV_LN2f;
  float s = 0.0f;
#pragma unroll
  for (int k = 0; k < 8; ++k) {
#pragma unroll
    for (int j = 0; j < 4; ++j)
      s += __builtin_amdgcn_exp2f(__builtin_fmaf(vv[k][j], INV_LN2f, -m2));
  }
  s = wave_red_sum(s);
  const float lse = __builtin_fmaf(__builtin_amdgcn_logf(s), LN2f, m); // m + ln2*log2(s)

  if (lane == 0) {
    lpB[row] = src[BLANKI] - lse;
    if (u < Uu) {
      const int lab = labels[b * Uu + u];
      lpL[(b * Tt + t) * Uu + u] = src[lab] - lse;
    }
  }
}

// -------- Phase 2: alpha recursion (1 wave per batch element) --------
__device__ __forceinline__ float logaddf(float a, float b) {
  float m = fmaxf(a, b);
  float d = fminf(a, b) - m;
  float z = __builtin_amdgcn_exp2f(d * INV_LN2f);
  return __builtin_fmaf(__builtin_amdgcn_logf(1.0f + z), LN2f, m);
}
__device__ __forceinline__ float scan_logadd(float x, int lane) {
#pragma unroll
  for (int d = 1; d < 32; d <<= 1) {
    float y = __shfl_up(x, d, 32);
    if (lane >= d) x = logaddf(x, y);
  }
  return x;
}
__device__ __forceinline__ float scan_add(float x, int lane) {
#pragma unroll
  for (int d = 1; d < 32; d <<= 1) {
    float y = __shfl_up(x, d, 32);
    if (lane >= d) x += y;
  }
  return x;
}

__global__ __launch_bounds__(32) void rnnt_alpha_kernel(
    const float* __restrict__ lpB, const float* __restrict__ lpL,
    const int* __restrict__ act_lens, const int* __restrict__ label_lens,
    float* __restrict__ out)
{
  const int b = blockIdx.x;
  const int lane = threadIdx.x;
  const int tEnd = act_lens[b] - 1;
  const int uIdx = label_lens[b];
  const float* __restrict__ Bp = lpB + (size_t)b * Tt * U1;
  const float* __restrict__ Lp = lpL + (size_t)b * Tt * Uu;

  // Lcum(t): exclusive add-scan of lp_label row; slots: u=lane, u=32+lane, u=64.
  float e0, e1, e64;
  {
    float i0 = scan_add(Lp[lane], lane);
    float tot0 = __shfl(i0, 31, 32);
    float i1 = scan_add(Lp[32 + lane], lane) + tot0;
    e64 = __shfl(i1, 31, 32);
    e0 = __shfl_up(i0, 1, 32); if (lane == 0) e0 = 0.0f;
    e1 = __shfl_up(i1, 1, 32); if (lane == 0) e1 = tot0;
  }
  float a0 = e0, a1 = e1, a64 = e64; // alpha0 = Lcum(:,0,:)

  for (int t = 1; t <= tEnd; ++t) {
    const float* Brow = Bp + (size_t)(t - 1) * U1;
    float bp0 = Brow[lane];
    float bp1 = Brow[32 + lane];
    float bp64 = Brow[64];
    const float* Lrow = Lp + (size_t)t * Uu;
    {
      float i0 = scan_add(Lrow[lane], lane);
      float tot0 = __shfl(i0, 31, 32);
      float i1 = scan_add(Lrow[32 + lane], lane) + tot0;
      e64 = __shfl(i1, 31, 32);
      e0 = __shfl_up(i0, 1, 32); if (lane == 0) e0 = 0.0f;
      e1 = __shfl_up(i1, 1, 32); if (lane == 0) e1 = tot0;
    }
    // x = alpha_prev + lp_blank(t-1) - Lcum(t); alpha_t = Lcum + cumlogsumexp(x)
    float x0 = a0 + bp0 - e0;
    float x1 = a1 + bp1 - e1;
    float x64 = a64 + bp64 - e64;
    float s0 = scan_logadd(x0, lane);
    float c0 = __shfl(s0, 31, 32);
    float s1 = logaddf(scan_logadd(x1, lane), c0);
    float c1 = __shfl(s1, 31, 32);
    float s64 = logaddf(c1, x64);
    a0 = e0 + s0;
    a1 = e1 + s1;
    a64 = e64 + s64;
  }

  float av;
  if (uIdx < 32)      av = __shfl(a0, uIdx, 32);
  else if (uIdx < 64) av = __shfl(a1, uIdx - 32, 32);
  else                av = a64;
  if (lane == 0)
    out[b] = -(av + Bp[(size_t)tEnd * U1 + uIdx]);
}

extern "C" void kernel_launch(void* const* d_in, const int* in_sizes, int n_in,
                              void* d_out, int out_size, void* d_ws, size_t ws_size,
                              hipStream_t stream) {
  const float* acts       = (const float*)d_in[0];
  const int*   labels     = (const int*)d_in[1];
  const int*   act_lens   = (const int*)d_in[2];
  const int*   label_lens = (const int*)d_in[3];
  float* out = (float*)d_out;
  float* lpB = (float*)d_ws;                       // B*T*(U+1) floats = 266,240 B
  float* lpL = lpB + (size_t)Bb * Tt * U1;         // B*T*U floats     = 262,144 B

  rnnt_rowlse_kernel<<<NROWS / ROWS_PER_BLOCK, 256, 0, stream>>>(acts, labels, lpB, lpL);
  rnnt_alpha_kernel<<<Bb, 32, 0, stream>>>(lpB, lpL, act_lens, label_lens, out);

  (void)in_sizes; (void)n_in; (void)out_size; (void)ws_size;
}